// ParametricPart_27779848471244
// MI455X (gfx1250) — compile-verified
//
#include <hip/hip_runtime.h>

#ifndef __has_builtin
#define __has_builtin(x) 0
#endif

#define AS1 __attribute__((address_space(1)))
#define AS3 __attribute__((address_space(3)))

typedef int v4i __attribute__((ext_vector_type(4)));

#if __has_builtin(__builtin_amdgcn_global_load_async_to_lds_b128)
#define HAVE_ASYNC_B128 1
#else
#define HAVE_ASYNC_B128 0
#endif

namespace {
constexpr int   Hh     = 256;
constexpr int   Ww     = 256;
constexpr int   ROWS   = 16;            // output rows per block
constexpr int   LROWS  = ROWS + 2;      // + halo
constexpr int   TPB    = 256;           // one thread per column (8 wave32s)
constexpr float INV_DX2 = 65536.0f;     // 1/(DX*DX), DX = 1/256 (exact pow2)
constexpr float INV_2DX = 128.0f;       // 1/(2*DX)

// ax[k] = (2 - |k/256*2 - 255/256|)/2 == 1 - |2k-255|/512 ; bit-exact in f32
__device__ __forceinline__ float axf(int i) {
  return 1.0f - fabsf(2.0f * (float)i - 255.0f) * (1.0f / 512.0f);
}
} // namespace

__global__ __launch_bounds__(TPB) void rd_stencil_kernel(
    const float* __restrict__ U, const float* __restrict__ d1p,
    const float* __restrict__ d2p, float* __restrict__ out)
{
  __shared__ float tile[LROWS * Ww];   // 18 KB

  const int    t    = threadIdx.x;                 // column 0..255
  const int    band = blockIdx.x % (Hh / ROWS);
  const int    img  = blockIdx.x / (Hh / ROWS);
  const int    r0   = band * ROWS;
  const size_t base = (size_t)img * (size_t)(Hh * Ww);
  const float* Ui   = U + base;

  // ---- Stage (ROWS+2) x 256 halo band into LDS with async global->LDS ----
  // 18 rows * 64 float4 = 1152 b128 transfers across 256 threads.
  for (int idx = t; idx < LROWS * (Ww / 4); idx += TPB) {
    const int lr = idx >> 6;                       // idx / (Ww/4)
    const int c4 = idx & 63;
    const int gr = (r0 - 1 + lr + Hh) & (Hh - 1);  // periodic row wrap
    const float* gp = Ui + (size_t)gr * Ww + (size_t)(c4 * 4);
    float*       lp = &tile[lr * Ww + c4 * 4];
#if HAVE_ASYNC_B128
    __builtin_amdgcn_global_load_async_to_lds_b128(
        (AS1 v4i*)(AS1 void*)(void*)const_cast<float*>(gp),
        (AS3 v4i*)(AS3 void*)(void*)lp, 0, 0);
#else
    *(float4*)lp = *(const float4*)gp;
#endif
  }

#if HAVE_ASYNC_B128
#if __has_builtin(__builtin_amdgcn_s_wait_asynccnt)
  __builtin_amdgcn_s_wait_asynccnt(0);
#else
  asm volatile("s_wait_asynccnt 0x0" ::: "memory");
#endif
#endif
  __syncthreads();

  const float D1 = *d1p;   // uniform scalar loads
  const float D2 = *d2p;

  const int   jm    = (t + Ww - 1) & (Ww - 1);
  const int   jp    = (t + 1) & (Ww - 1);
  const float ax_jm = axf(jm);
  const float ax_jp = axf(jp);

#pragma unroll
  for (int rr = 0; rr < ROWS; ++rr) {
    const int   gi    = r0 + rr;
    const float axr_m = axf((gi - 1 + Hh) & (Hh - 1));
    const float axr_p = axf((gi + 1) & (Hh - 1));

    const float n = tile[(rr + 0) * Ww + t];   // row gi-1
    const float c = tile[(rr + 1) * Ww + t];   // row gi
    const float s = tile[(rr + 2) * Ww + t];   // row gi+1
    const float w = tile[(rr + 1) * Ww + jm];  // col j-1
    const float e = tile[(rr + 1) * Ww + jp];  // col j+1

    // Laplacian / DX^2
    const float lap = (n + s + w + e - 4.0f * c) * INV_DX2;

    // EU_x at row neighbors (axis_x = ax[row]); EU_y at col neighbors
    const float eux_p = s * fmaf(-s, axr_p, 1.0f);
    const float eux_m = n * fmaf(-n, axr_m, 1.0f);
    const float euy_p = e * fmaf(-e, ax_jp, 1.0f);
    const float euy_m = w * fmaf(-w, ax_jm, 1.0f);
    const float dv = ((eux_p - eux_m) + (euy_p - euy_m)) * INV_2DX;

    const float r = fmaf(D1, lap, D2 * dv);
    // NT store: keep `state` resident in the 192 MB L2 for row reuse
    __builtin_nontemporal_store(r, &out[base + (size_t)gi * Ww + t]);
  }
}

extern "C" void kernel_launch(void* const* d_in, const int* in_sizes, int n_in,
                              void* d_out, int out_size, void* d_ws, size_t ws_size,
                              hipStream_t stream) {
  (void)n_in; (void)out_size; (void)d_ws; (void)ws_size;
  const float* U  = (const float*)d_in[0];
  const float* D1 = (const float*)d_in[1];
  const float* D2 = (const float*)d_in[2];
  float* out = (float*)d_out;

  const int nimg   = in_sizes[0] / (Hh * Ww);       // e.g. 512 images
  const int blocks = nimg * (Hh / ROWS);            // 8192 blocks
  rd_stencil_kernel<<<blocks, TPB, 0, stream>>>(U, D1, D2, out);
}